// GNNClassifier_88648124990589
// MI455X (gfx1250) — compile-verified
//
#include <hip/hip_runtime.h>
#include <hip/hip_bf16.h>

// ---------------------------------------------------------------------------
// GNN (SAGEConv x2 + BN + ReLU + mean-pool + linear head) for MI455X/gfx1250.
//
// Roofline: edge gather/scatter dominates (~1 GB traffic -> ~50us @ 23.3TB/s);
// done with float4 gathers + global_atomic_add_f32. Dense SAGE linear layers
// (~2.7 GFLOP fp32) use V_WMMA_F32_16X16X4_F32 with weights staged in LDS in
// a K-pair-interleaved layout so every B fragment is a single aligned
// ds_load_b64 straight into an even VGPR pair (no repack movs).
// ---------------------------------------------------------------------------

typedef float v2f __attribute__((ext_vector_type(2)));
typedef float v8f __attribute__((ext_vector_type(8)));

#define HID 64
#define INF0 40
#define EPSV 1e-5f

// ---------------- generic zero ----------------
__global__ void k_zero(float* __restrict__ p, long n) {
    long i = (long)blockIdx.x * blockDim.x + threadIdx.x;
    if (i < n) p[i] = 0.0f;
}

// ---------------- embedding concat: x0[n,40] ----------------
__global__ void k_embed(const int* __restrict__ sid, const int* __restrict__ cid,
                        const int* __restrict__ pid,
                        const float* __restrict__ stab, const float* __restrict__ ctab,
                        const float* __restrict__ ptab,
                        float* __restrict__ x0, int n_nodes) {
    long i = (long)blockIdx.x * blockDim.x + threadIdx.x;
    long total = (long)n_nodes * INF0;
    if (i >= total) return;
    int n = (int)(i / INF0);
    int f = (int)(i % INF0);
    float v;
    if (f < 16)       v = stab[sid[n] * 16 + f];
    else if (f < 32)  v = ctab[cid[n] * 16 + (f - 16)];
    else              v = ptab[pid[n] * 8 + (f - 32)];
    x0[i] = v;
}

// ---------------- degree + reciprocal ----------------
__global__ void k_deg(const int* __restrict__ dst, float* __restrict__ deg, int nE) {
    long e = (long)blockIdx.x * blockDim.x + threadIdx.x;
    if (e < nE) atomicAdd(&deg[dst[e]], 1.0f);
}
__global__ void k_invdeg(float* __restrict__ deg, int n_nodes) {
    long i = (long)blockIdx.x * blockDim.x + threadIdx.x;
    if (i < n_nodes) deg[i] = 1.0f / fmaxf(deg[i], 1.0f);
}

// ---------------- edge scatter (float4 chunks): agg[dst,c:c+4] += x[src,c:c+4]
__global__ void k_scatter4(const int* __restrict__ src, const int* __restrict__ dst,
                           const float* __restrict__ x, int ldx,
                           float* __restrict__ agg, int nE, int F4) {
    long i = (long)blockIdx.x * blockDim.x + threadIdx.x;
    long total = (long)nE * F4;
    if (i >= total) return;
    int e = (int)(i / F4);
    int c = (int)(i % F4) * 4;
    int s = src[e], d = dst[e];
    const float4 v = *(const float4*)(x + (long)s * ldx + c);
    float* p = agg + (long)d * HID + c;
    atomicAdd(p + 0, v.x);
    atomicAdd(p + 1, v.y);
    atomicAdd(p + 2, v.z);
    atomicAdd(p + 3, v.w);
}

// ---------------- fused SAGE GEMM via fp32 WMMA -----------------------------
// h[n,0:64] = (agg[n,0:K]*invdeg[n]) @ Wl + x[n,0:K] @ Wr + bias
// One wave -> 16(M) x 64(N) output tile, 4 v8f accumulators.
// LDS weight layout (K-pair interleaved):
//   sW[kp*128 + col*2 + r] = W[(2*kp + r)*64 + col]   (kp = k/2, r = k&1)
// so lane (col=l%16+16j, kpair=(k>>1)+(l>>4)) reads its B frag as one
// contiguous 8B ds_load_b64.
// A frag (16x4 f32): lane l holds row l%16, K pair (l/16)*2.
// C/D (16x16 f32): vgpr g -> row g + (l/16)*8, col l%16.
__global__ __launch_bounds__(128)
void k_sage_gemm(const float* __restrict__ agg, const float* __restrict__ invdeg,
                 const float* __restrict__ x, int ldx,
                 const float* __restrict__ Wl, const float* __restrict__ Wr,
                 const float* __restrict__ bias, int K,
                 float* __restrict__ h, int n_nodes) {
    extern __shared__ float smem[];              // [K*64 | K*64 | 64]
    float* sWl = smem;
    float* sWr = smem + (long)K * HID;
    float* sb  = smem + 2L * K * HID;

    for (int idx = threadIdx.x; idx < K * HID; idx += blockDim.x) {
        int k   = idx / HID;
        int col = idx % HID;
        int off = (k >> 1) * 128 + col * 2 + (k & 1);
        sWl[off] = Wl[idx];
        sWr[off] = Wr[idx];
    }
    if (threadIdx.x < HID) sb[threadIdx.x] = bias[threadIdx.x];
    __syncthreads();

    int wave = blockIdx.x * (blockDim.x >> 5) + (threadIdx.x >> 5);
    int lane = threadIdx.x & 31;
    int row0 = wave * 16;
    if (row0 >= n_nodes) return;                 // wave-uniform (grid is exact)
    int rsel  = lane & 15;                       // A: row-in-tile, B: col-in-tile
    int hi    = lane >> 4;                       // 0 or 1
    int khalf = hi * 2;                          // 0 or 2
    int arow  = row0 + rsel;
    float idg = invdeg[arow];

    v8f c[4];
    #pragma unroll
    for (int j = 0; j < 4; ++j) c[j] = (v8f){0.f,0.f,0.f,0.f,0.f,0.f,0.f,0.f};

    // ---- neighbor-mean half: (agg * invdeg) @ Wl ----
    for (int k = 0; k < K; k += 4) {
        v2f a;
        a.x = agg[(long)arow * HID + k + khalf]     * idg;
        a.y = agg[(long)arow * HID + k + khalf + 1] * idg;
        const float* W = sWl + ((k >> 1) + hi) * 128;
        #pragma unroll
        for (int j = 0; j < 4; ++j) {
            v2f b = *(const v2f*)(W + (j * 16 + rsel) * 2);
            c[j] = __builtin_amdgcn_wmma_f32_16x16x4_f32(
                false, a, false, b, (short)0, c[j], false, false);
        }
    }
    // ---- self half: x @ Wr ----
    for (int k = 0; k < K; k += 4) {
        v2f a;
        a.x = x[(long)arow * ldx + k + khalf];
        a.y = x[(long)arow * ldx + k + khalf + 1];
        const float* W = sWr + ((k >> 1) + hi) * 128;
        #pragma unroll
        for (int j = 0; j < 4; ++j) {
            v2f b = *(const v2f*)(W + (j * 16 + rsel) * 2);
            c[j] = __builtin_amdgcn_wmma_f32_16x16x4_f32(
                false, a, false, b, (short)0, c[j], false, false);
        }
    }

    int rbase = row0 + hi * 8;
    #pragma unroll
    for (int j = 0; j < 4; ++j) {
        float bv = sb[j * 16 + rsel];
        #pragma unroll
        for (int g = 0; g < 8; ++g)
            h[(long)(rbase + g) * HID + j * 16 + rsel] = c[j][g] + bv;
    }
}

// ---------------- BatchNorm stats: per-column sum & sumsq -------------------
__global__ void k_colreduce(const float* __restrict__ h, int n_nodes,
                            float* __restrict__ stats /*[128]: sum, sumsq*/) {
    int f   = threadIdx.x & 63;
    int sub = threadIdx.x >> 6;                  // 0..3
    int r0  = blockIdx.x * 512 + sub;
    float s = 0.f, q = 0.f;
    for (int i = 0; i < 128; ++i) {
        int r = r0 + i * 4;
        if (r < n_nodes) {
            float v = h[(long)r * HID + f];
            s += v; q += v * v;
        }
    }
    atomicAdd(&stats[f], s);
    atomicAdd(&stats[64 + f], q);
}

__global__ void k_bnparams(const float* __restrict__ stats,
                           const float* __restrict__ gamma,
                           const float* __restrict__ beta,
                           float* __restrict__ sc /*[128]: scale, shift*/,
                           int n_nodes) {
    int f = threadIdx.x;
    if (f >= HID) return;
    float invn = 1.0f / (float)n_nodes;
    float m = stats[f] * invn;
    float v = stats[64 + f] * invn - m * m;
    float s = gamma[f] * rsqrtf(v + EPSV);
    sc[f]      = s;
    sc[64 + f] = beta[f] - m * s;
}

__global__ void k_bnrelu(float* __restrict__ h, const float* __restrict__ sc, long total) {
    long i = (long)blockIdx.x * blockDim.x + threadIdx.x;
    if (i >= total) return;
    int f = (int)(i & 63);
    float v = h[i] * sc[f] + sc[64 + f];
    h[i] = fmaxf(v, 0.0f);
}

// ---------------- global mean pool + head -----------------------------------
__global__ void k_poolsum4(const float* __restrict__ x, const int* __restrict__ batch,
                           float* __restrict__ pooled, int n_nodes) {
    long i = (long)blockIdx.x * blockDim.x + threadIdx.x;
    long total = (long)n_nodes * (HID / 4);
    if (i >= total) return;
    int n = (int)(i / (HID / 4));
    int c = (int)(i % (HID / 4)) * 4;
    const float4 v = *(const float4*)(x + (long)n * HID + c);
    float* p = pooled + (long)batch[n] * HID + c;
    atomicAdd(p + 0, v.x);
    atomicAdd(p + 1, v.y);
    atomicAdd(p + 2, v.z);
    atomicAdd(p + 3, v.w);
}
__global__ void k_poolcnt(const int* __restrict__ batch, float* __restrict__ cnt, int n_nodes) {
    long i = (long)blockIdx.x * blockDim.x + threadIdx.x;
    if (i < n_nodes) atomicAdd(&cnt[batch[i]], 1.0f);
}
__global__ void k_head(const float* __restrict__ pooled, const float* __restrict__ cnt,
                       const float* __restrict__ Wout, const float* __restrict__ bout,
                       float* __restrict__ out, int G) {
    int g = blockIdx.x * blockDim.x + threadIdx.x;
    if (g >= G) return;
    float inv = 1.0f / fmaxf(cnt[g], 1.0f);
    float a0 = 0.f, a1 = 0.f;
    for (int f = 0; f < HID; ++f) {
        float p = pooled[(long)g * HID + f] * inv;
        a0 += p * Wout[f * 2 + 0];
        a1 += p * Wout[f * 2 + 1];
    }
    out[g * 2 + 0] = a0 + bout[0];
    out[g * 2 + 1] = a1 + bout[1];
}

// ---------------------------------------------------------------------------
static inline int gdim(long n, int b) { return (int)((n + b - 1) / b); }

extern "C" void kernel_launch(void* const* d_in, const int* in_sizes, int n_in,
                              void* d_out, int out_size, void* d_ws, size_t ws_size,
                              hipStream_t stream) {
    const int*   sid   = (const int*)d_in[0];
    const int*   cid   = (const int*)d_in[1];
    const int*   pid   = (const int*)d_in[2];
    const int*   eidx  = (const int*)d_in[3];
    const int*   batch = (const int*)d_in[4];
    const float* stab  = (const float*)d_in[6];
    const float* ctab  = (const float*)d_in[7];
    const float* ptab  = (const float*)d_in[8];
    const float* W1l   = (const float*)d_in[9];
    const float* b1    = (const float*)d_in[10];
    const float* W1r   = (const float*)d_in[11];
    const float* g1    = (const float*)d_in[12];
    const float* be1   = (const float*)d_in[13];
    const float* W2l   = (const float*)d_in[14];
    const float* b2    = (const float*)d_in[15];
    const float* W2r   = (const float*)d_in[16];
    const float* g2    = (const float*)d_in[17];
    const float* be2   = (const float*)d_in[18];
    const float* Wout  = (const float*)d_in[19];
    const float* bout  = (const float*)d_in[20];
    float* out = (float*)d_out;

    const int N = in_sizes[0];
    const int E = in_sizes[3] / 2;
    const int G = out_size / 2;
    const int* src = eidx;
    const int* dst = eidx + E;

    // workspace layout (floats); total ~= 233*N + 256 + 65*G  (~75 MB)
    float* ws      = (float*)d_ws;
    float* x0      = ws;                       // N*40
    float* invdeg  = x0 + (long)N * INF0;      // N
    float* agg     = invdeg + N;               // N*64
    float* h1      = agg + (long)N * HID;      // N*64
    float* h2      = h1 + (long)N * HID;       // N*64
    float* stats   = h2 + (long)N * HID;       // 128
    float* bnsc    = stats + 128;              // 128
    float* pooled  = bnsc + 128;               // G*64
    float* cnt     = pooled + (long)G * HID;   // G
    (void)ws_size; (void)n_in;

    const int B = 256;
    const size_t lds1 = (size_t)(2 * INF0 * HID + HID) * sizeof(float); // ~20.5 KB
    const size_t lds2 = (size_t)(2 * HID  * HID + HID) * sizeof(float); // ~32.25 KB

    // features + degrees
    k_embed<<<gdim((long)N * INF0, B), B, 0, stream>>>(sid, cid, pid, stab, ctab, ptab, x0, N);
    k_zero <<<gdim(N, B), B, 0, stream>>>(invdeg, N);
    k_deg  <<<gdim(E, B), B, 0, stream>>>(dst, invdeg, E);
    k_invdeg<<<gdim(N, B), B, 0, stream>>>(invdeg, N);

    // ---- layer 1 ----
    k_zero    <<<gdim((long)N * HID, B), B, 0, stream>>>(agg, (long)N * HID);
    k_scatter4<<<gdim((long)E * (INF0 / 4), B), B, 0, stream>>>(src, dst, x0, INF0, agg, E, INF0 / 4);
    k_sage_gemm<<<gdim((long)(N / 16), 4), 128, lds1, stream>>>(agg, invdeg, x0, INF0,
                                                                W1l, W1r, b1, INF0, h1, N);
    k_zero     <<<1, 128, 0, stream>>>(stats, 128);
    k_colreduce<<<gdim(N, 512), 256, 0, stream>>>(h1, N, stats);
    k_bnparams <<<1, 64, 0, stream>>>(stats, g1, be1, bnsc, N);
    k_bnrelu   <<<gdim((long)N * HID, B), B, 0, stream>>>(h1, bnsc, (long)N * HID);

    // ---- layer 2 ----
    k_zero    <<<gdim((long)N * HID, B), B, 0, stream>>>(agg, (long)N * HID);
    k_scatter4<<<gdim((long)E * (HID / 4), B), B, 0, stream>>>(src, dst, h1, HID, agg, E, HID / 4);
    k_sage_gemm<<<gdim((long)(N / 16), 4), 128, lds2, stream>>>(agg, invdeg, h1, HID,
                                                                W2l, W2r, b2, HID, h2, N);
    k_zero     <<<1, 128, 0, stream>>>(stats, 128);
    k_colreduce<<<gdim(N, 512), 256, 0, stream>>>(h2, N, stats);
    k_bnparams <<<1, 64, 0, stream>>>(stats, g2, be2, bnsc, N);
    k_bnrelu   <<<gdim((long)N * HID, B), B, 0, stream>>>(h2, bnsc, (long)N * HID);

    // ---- pool + head ----
    k_zero    <<<gdim((long)G * HID + G, B), B, 0, stream>>>(pooled, (long)G * HID + G);
    k_poolsum4<<<gdim((long)N * (HID / 4), B), B, 0, stream>>>(h2, batch, pooled, N);
    k_poolcnt <<<gdim(N, B), B, 0, stream>>>(batch, cnt, N);
    k_head    <<<gdim(G, B), B, 0, stream>>>(pooled, cnt, Wout, bout, out, G);
}